// EvolutionGeneratorLognormal_36593121362479
// MI455X (gfx1250) — compile-verified
//
#include <hip/hip_runtime.h>

// CDNA5 / gfx1250, wave32.
typedef __attribute__((ext_vector_type(2))) float v2f;
typedef __attribute__((ext_vector_type(4))) float v4f;
typedef __attribute__((ext_vector_type(8))) float v8f;

#define N_DATES  64
#define M_ASSETS 64
#define N_SIMS   32768

// covT: row stride 66 floats -> A-frag row gather hits bank 2*row (conflict-free,
//       and 8B-aligned b64 loads since 66*row + 4*kk is even).
#define COV_STRIDE 66
// zT: K-pair interleaved layout. Element (k, s) at word (k>>1)*ZP_STRIDE + 2*s + (k&1).
//     A B-frag {K, K+1} is then ONE contiguous ds_load_b64 (no register packing).
//     ZP_STRIDE = 160: 160 mod 64 == 32, so the hi half-wave (pair 2kk+1) uses the
//     other 32 LDS banks than the lo half-wave (pair 2kk) -> conflict-free b64 wave load.
#define ZP_STRIDE 160   // words per k-pair row (64 cols * 2 + 32 pad)

__global__ __launch_bounds__(128) void evo_lognormal_kernel(
    const float* __restrict__ cov,   // [64, 64, 64]
    const float* __restrict__ var,   // [64, 64, 1]
    const float* __restrict__ z,     // [64, 64, 32768]
    float* __restrict__ out)         // [64, 64, 32768]
{
    __shared__ __align__(16) float covT[64 * COV_STRIDE];
    __shared__ __align__(16) float zT[32 * ZP_STRIDE];   // 20 KB
    __shared__ float varT[64];

    const int tid    = threadIdx.x;
    const int wave   = tid >> 5;          // 4 waves of 32
    const int lane   = tid & 31;
    const int lrow   = lane & 15;
    const bool hi    = lane >= 16;
    const int m_base = wave * 16;         // each wave owns 16 output rows
    const int s_base = blockIdx.x * 64;   // each block owns 64 sim columns

    // Running cumulative sum lives in the WMMA C registers: 4 subtiles of 16x16.
    v8f acc[4];
#pragma unroll
    for (int st = 0; st < 4; ++st)
#pragma unroll
        for (int r = 0; r < 8; ++r) acc[st][r] = 0.0f;

    for (int n = 0; n < N_DATES; ++n) {
        __syncthreads();

        // ---- Stage cov[n] (64x64 f32 = 16KB) into LDS, coalesced float2s.
        {
            const float* covn = cov + (size_t)n * 64 * 64;
#pragma unroll
            for (int i = 0; i < 16; ++i) {
                int idx = i * 128 + tid;          // 2048 float2 total
                int row = idx >> 5;               // 32 float2 per row
                int c2  = (idx & 31) * 2;
                v2f d = *(const v2f*)(covn + row * 64 + c2);
                *(v2f*)(&covT[row * COV_STRIDE + c2]) = d;
            }
        }
        // ---- Stage z[n][:, s_base:s_base+64] into LDS in K-pair-interleaved form.
        //      Thread unit: rows {2p, 2p+1} x cols {2c2, 2c2+1}:
        //      two coalesced global float2 loads -> one 16B ds_store_b128.
        {
            const float* zn = z + (size_t)n * 64 * N_SIMS + s_base;
#pragma unroll
            for (int i = 0; i < 8; ++i) {
                int idx = i * 128 + tid;          // 1024 units
                int p   = idx >> 5;               // k-pair index 0..31
                int c2  = idx & 31;               // float2 column index
                const float* g0 = zn + (size_t)(2 * p) * N_SIMS + 2 * c2;
                v2f d0 = *(const v2f*)g0;             // row 2p
                v2f d1 = *(const v2f*)(g0 + N_SIMS);  // row 2p+1
                v4f w;
                w.x = d0.x; w.y = d1.x; w.z = d0.y; w.w = d1.y;
                *(v4f*)(&zT[p * ZP_STRIDE + 4 * c2]) = w;
            }
        }
        if (tid < 64) varT[tid] = var[n * 64 + tid];

        // Prefetch next date's z slab rows into cache while we compute.
        if (n + 1 < N_DATES) {
            const float* znext = z + (size_t)(n + 1) * 64 * N_SIMS + s_base;
            __builtin_prefetch(znext + (size_t)(tid & 63) * N_SIMS, 0, 3);
        }
        __syncthreads();

        // ---- acc += cov[n][m_base:m_base+16, :] @ z_slab   (16x64x64 via f32 WMMA)
        // A-frag (16x4 f32): lanes 0-15 hold K={0,1}, lanes 16-31 hold K={2,3}.
        // B-frag (4x16 f32): VGPR v, half h -> K = 4*kk + 2*h + v; col = lane%16.
        const int pbase = hi ? 1 : 0;         // k-pair offset for this half-wave
        const int koff  = hi ? 2 : 0;
#pragma unroll
        for (int kk = 0; kk < 16; ++kk) {
            v2f a = *(const v2f*)(&covT[(m_base + lrow) * COV_STRIDE + 4 * kk + koff]);
            const float* brow = &zT[(2 * kk + pbase) * ZP_STRIDE + 2 * lrow];
#pragma unroll
            for (int st = 0; st < 4; ++st) {
                v2f b = *(const v2f*)(brow + st * 32);   // {z[K][s], z[K+1][s]} contiguous
                // (neg_a, A, neg_b, B, c_mod, C, reuse_a, reuse_b)
                acc[st] = __builtin_amdgcn_wmma_f32_16x16x4_f32(
                    false, a, false, b, (short)0, acc[st], false, false);
            }
        }

        // ---- Fold -var[n]/2 into the running sum (cumsum applies it once/date),
        //      then store this date's cumulative tile.
        // C layout: VGPR r, lanes 0-15 -> M = m_base + r; lanes 16-31 -> M = m_base + 8 + r.
        const int mrow0 = m_base + (hi ? 8 : 0);
#pragma unroll
        for (int r = 0; r < 8; ++r) {
            float bias = 0.5f * varT[mrow0 + r];
#pragma unroll
            for (int st = 0; st < 4; ++st) acc[st][r] -= bias;
        }

        float* outn = out + (size_t)n * 64 * N_SIMS + s_base;
#pragma unroll
        for (int st = 0; st < 4; ++st)
#pragma unroll
            for (int r = 0; r < 8; ++r)
                outn[(size_t)(mrow0 + r) * N_SIMS + st * 16 + lrow] = acc[st][r];
    }
}

extern "C" void kernel_launch(void* const* d_in, const int* in_sizes, int n_in,
                              void* d_out, int out_size, void* d_ws, size_t ws_size,
                              hipStream_t stream) {
    const float* cov = (const float*)d_in[0];   // [64,64,64]
    const float* var = (const float*)d_in[1];   // [64,64,1]
    const float* z   = (const float*)d_in[2];   // [64,64,32768]
    float* out       = (float*)d_out;           // [64,64,32768]

    dim3 grid(N_SIMS / 64);   // 512 blocks, each owns a 64-wide sim slab
    dim3 block(128);          // 4 wave32s: one 16-row m-tile per wave
    evo_lognormal_kernel<<<grid, block, 0, stream>>>(cov, var, z, out);
}